// MyGNN_52501680226567
// MI455X (gfx1250) — compile-verified
//
#include <hip/hip_runtime.h>
#include <hip/hip_bf16.h>
#include <math.h>

// GATConv on complete "i<j + self loop" graph == causal attention with
// rank-1 scores: out[j] = relu( sum_{i<=j} softmax_i(lrelu(a_s[i]+a_d[j])) * h[i] + bias )
//
// N=2048, Cin=128, Cout=64, fp32. Footprint fits in L2 (192MB), so fp32 WMMA
// (V_WMMA_F32_16X16X4_F32) is the right precision: matches reference numerics
// and keeps all MACs on the matrix pipe. h is materialized directly in the
// WMMA B-register layout so the hot attention loop loads B operands as single
// coalesced b64s.

#define NN    2048
#define CIN   128
#define COUT  64
#define NEG_SLOPE 0.2f

typedef float v2f __attribute__((ext_vector_type(2)));
typedef float v8f __attribute__((ext_vector_type(8)));

// packed-h indexing: hp[((s*4 + chunk)*4 + colgrp)*32 + lane] (float2)
//   lane L: col = colgrp*16 + (L&15), rows = s*16 + 4*chunk + 2*(L>>4) + {0,1}
// one src block s = 512 float2 = 4KB, contiguous.

// ---------------------------------------------------------------------------
// Kernel 1: h = data @ W (16 rows per wave, WMMA f32 16x16x4, K=128).
// Emits h in packed B layout + a_s = h.att_src, a_d = h.att_dst.
// A layout (ISA 7.12.2, 32-bit A 16x4): lane L holds row M=L%16,
//   VGPR0 = K + 2*(L>>4), VGPR1 = +1. B: same K striping, col on lanes.
// C/D layout: VGPR v, lanes 0-15 -> M=v, lanes 16-31 -> M=v+8, N=L%16.
// ---------------------------------------------------------------------------
__global__ __launch_bounds__(32)
void gat_h_as_ad(const float* __restrict__ data, const float* __restrict__ W,
                 const float* __restrict__ att_src, const float* __restrict__ att_dst,
                 v2f* __restrict__ hp, float* __restrict__ a_s, float* __restrict__ a_d)
{
    __shared__ float sh[16 * COUT];

    const int lane = threadIdx.x;          // wave32
    const int ml   = lane & 15;
    const int hi   = lane >> 4;            // 0 or 1
    const int kh   = hi * 2;
    const int row0 = blockIdx.x * 16;

    v8f acc0 = {}, acc1 = {}, acc2 = {}, acc3 = {};

    for (int k = 0; k < CIN; k += 4) {
        const float* arow = data + (size_t)(row0 + ml) * CIN + k + kh;
        v2f a; a.x = arow[0]; a.y = arow[1];

        const float* brow0 = W + (size_t)(k + kh) * COUT;
        const float* brow1 = brow0 + COUT;

        v2f b0; b0.x = brow0[ 0 + ml]; b0.y = brow1[ 0 + ml];
        v2f b1; b1.x = brow0[16 + ml]; b1.y = brow1[16 + ml];
        v2f b2; b2.x = brow0[32 + ml]; b2.y = brow1[32 + ml];
        v2f b3; b3.x = brow0[48 + ml]; b3.y = brow1[48 + ml];

        acc0 = __builtin_amdgcn_wmma_f32_16x16x4_f32(false, a, false, b0, (short)0, acc0, false, false);
        acc1 = __builtin_amdgcn_wmma_f32_16x16x4_f32(false, a, false, b1, (short)0, acc1, false, false);
        acc2 = __builtin_amdgcn_wmma_f32_16x16x4_f32(false, a, false, b2, (short)0, acc2, false, false);
        acc3 = __builtin_amdgcn_wmma_f32_16x16x4_f32(false, a, false, b3, (short)0, acc3, false, false);
    }

    // stage C tile in LDS (row-major within tile)
    #pragma unroll
    for (int v = 0; v < 8; ++v) {
        const int m = v + hi * 8;
        float* lr = sh + m * COUT;
        lr[ 0 + ml] = acc0[v];
        lr[16 + ml] = acc1[v];
        lr[32 + ml] = acc2[v];
        lr[48 + ml] = acc3[v];
    }
    __syncthreads();

    // repack into WMMA-B layout: each lane emits 16 float2 (whole 16x64 tile)
    #pragma unroll
    for (int t = 0; t < 16; ++t) {
        const int c  = t >> 2;          // chunk 0..3 (K group of 4)
        const int wt = t & 3;           // column group 0..3
        const int kr = 4 * c + kh;      // local row pair base
        const int cl = wt * 16 + ml;    // column
        v2f v; v.x = sh[kr * COUT + cl]; v.y = sh[(kr + 1) * COUT + cl];
        hp[((size_t)(blockIdx.x * 4 + c) * 4 + wt) * 32 + lane] = v;
    }

    // lanes 0-15: a_s for row ml ; lanes 16-31: a_d for row ml
    const float* av = (hi == 0) ? att_src : att_dst;
    float s = 0.f;
    #pragma unroll 8
    for (int c = 0; c < COUT; ++c) s += sh[ml * COUT + c] * av[c];
    if (hi == 0) a_s[row0 + ml] = s;
    else         a_d[row0 + ml] = s;
}

// ---------------------------------------------------------------------------
// Kernel 2: flash-style causal attention. One block = one 16-dst-row tile;
// 4 waves, wave w owns output columns [16w, 16w+16). Scores are rank-1
// (a_s[i]+a_d[j]) built directly in the WMMA A-register layout.
// ---------------------------------------------------------------------------
__global__ __launch_bounds__(128)
void gat_attn(const v2f* __restrict__ hp, const float* __restrict__ a_s,
              const float* __restrict__ a_d, const float* __restrict__ bias,
              float* __restrict__ out)
{
    __shared__ float s_as[NN];

    const int tid  = threadIdx.x;
    const int wid  = tid >> 5;
    const int lane = tid & 31;
    const int ml   = lane & 15;
    const int hi   = lane >> 4;
    const int kh   = hi * 2;

    // heaviest tiles (largest d) launch first
    const int d    = (int)gridDim.x - 1 - (int)blockIdx.x;
    const int nblk = d + 1;                      // src blocks 0..d

    for (int t = tid; t < nblk * 16; t += 128) s_as[t] = a_s[t];
    __syncthreads();

    const int   j    = d * 16 + ml;              // dst row this lane tracks stats for
    const float adj  = a_d[j];
    const int   ncol = wid * 16 + ml;            // output column this lane owns in O

    float mrun = -INFINITY;
    float lrun = 0.f;
    v8f   acc  = {};

    // per-wave base into packed h for our column group
    const v2f* hpw = hp + (size_t)wid * 32 + lane;

    for (int s = 0; s < nblk; ++s) {
        if (s + 1 < nblk) {
            // warm next packed tile (4KB contiguous): 32 lanes x 128B stride
            __builtin_prefetch((const float*)(hp + (size_t)(s + 1) * 512) + lane * 32, 0, 3);
        }

        // --- scores in A-register layout: lane holds row ml, K pairs {4c+kh, +1}
        float p[8];
        float rmax = -INFINITY;
        #pragma unroll
        for (int c = 0; c < 4; ++c) {
            const int kbase = s * 16 + 4 * c + kh;
            const v2f asv = *(const v2f*)&s_as[kbase];     // one ds_load_b64
            #pragma unroll
            for (int u = 0; u < 2; ++u) {
                const int i = kbase + u;                   // global src
                float e = (u ? asv.y : asv.x) + adj;
                e = (e > 0.f) ? e : NEG_SLOPE * e;         // leaky_relu
                if (i > j) e = -INFINITY;                  // causal mask (only s==d)
                p[c * 2 + u] = e;
                rmax = fmaxf(rmax, e);
            }
        }
        rmax = fmaxf(rmax, __shfl_xor(rmax, 16));          // other half-wave's 8 Ks

        const float mnew = fmaxf(mrun, rmax);
        const float sc   = __expf(mrun - mnew);            // 0 on first iteration
        mrun = mnew;

        float rsum = 0.f;
        #pragma unroll
        for (int q = 0; q < 8; ++q) { p[q] = __expf(p[q] - mnew); rsum += p[q]; }
        rsum += __shfl_xor(rsum, 16);
        lrun = lrun * sc + rsum;

        // --- rescale O (row M = v + 8*hi needs scale of stat-lane M)
        #pragma unroll
        for (int v = 0; v < 8; ++v) acc[v] *= __shfl(sc, v + hi * 8);

        // --- O += P_exp @ h_block : 4 chained wmma, B = one coalesced b64 each
        #pragma unroll
        for (int c = 0; c < 4; ++c) {
            v2f b = hpw[((size_t)s * 4 + c) * 4 * 32];
            v2f a; a.x = p[2 * c]; a.y = p[2 * c + 1];
            acc = __builtin_amdgcn_wmma_f32_16x16x4_f32(false, a, false, b, (short)0, acc, false, false);
        }
    }

    const float bcol = bias[ncol];
    #pragma unroll
    for (int v = 0; v < 8; ++v) {
        const int m = v + hi * 8;
        const float linv = 1.0f / __shfl(lrun, m);
        float o = acc[v] * linv + bcol;
        o = fmaxf(o, 0.f);                                 // final relu
        out[(size_t)(d * 16 + m) * COUT + ncol] = o;
    }
}

// ---------------------------------------------------------------------------
extern "C" void kernel_launch(void* const* d_in, const int* in_sizes, int n_in,
                              void* d_out, int out_size, void* d_ws, size_t ws_size,
                              hipStream_t stream) {
    const float* data    = (const float*)d_in[0];   // [2048,128]
    const float* W       = (const float*)d_in[1];   // [128,64]
    const float* att_src = (const float*)d_in[2];   // [64]
    const float* att_dst = (const float*)d_in[3];   // [64]
    const float* bias    = (const float*)d_in[4];   // [64]
    float* out = (float*)d_out;                     // [2048,64]

    // scratch: hp (2048*64 floats as float2, B-layout) | a_s (2048) | a_d (2048)
    v2f*   hp  = (v2f*)d_ws;
    float* a_s = (float*)d_ws + (size_t)NN * COUT;
    float* a_d = a_s + NN;

    gat_h_as_ad<<<NN / 16, 32, 0, stream>>>(data, W, att_src, att_dst, hp, a_s, a_d);
    gat_attn  <<<NN / 16, 128, 0, stream>>>(hp, a_s, a_d, bias, out);
}